// pointnet_sa_module_87969520157198
// MI455X (gfx1250) — compile-verified
//
#include <hip/hip_runtime.h>

#define BATCH   8
#define NPTS    4096
#define CIN     64
#define NPOINT  1024
#define NSAMPLE 32

typedef __attribute__((ext_vector_type(2))) float v2f;
typedef __attribute__((ext_vector_type(8))) float v8f;

// ---------------------------------------------------------------------------
// Kernel 1: Farthest point sampling. One workgroup (256 thr) per batch.
// xyz SoA + running min-dist kept in LDS; 1024 sequential argmax steps.
// ---------------------------------------------------------------------------
__global__ void fps_kernel(const float* __restrict__ xyz, float* __restrict__ new_xyz)
{
    extern __shared__ float sm[];
    float* px   = sm;
    float* py   = sm + NPTS;
    float* pz   = sm + 2 * NPTS;
    float* dist = sm + 3 * NPTS;
    float* redV = sm + 4 * NPTS;          // 8 floats
    int*   redI = (int*)(sm + 4 * NPTS + 8);
    int*   farS = (int*)(sm + 4 * NPTS + 16);

    const int b   = blockIdx.x;
    const int tid = threadIdx.x;
    const float* X = xyz + (size_t)b * NPTS * 3;

    for (int i = tid; i < NPTS; i += 256) {
        px[i]   = X[i * 3 + 0];
        py[i]   = X[i * 3 + 1];
        pz[i]   = X[i * 3 + 2];
        dist[i] = 1e10f;
    }
    __syncthreads();

    int far = 0;
    float* out = new_xyz + (size_t)b * NPOINT * 3;
    for (int it = 0; it < NPOINT; ++it) {
        const float cx = px[far], cy = py[far], cz = pz[far];
        if (tid == 0) { out[it * 3 + 0] = cx; out[it * 3 + 1] = cy; out[it * 3 + 2] = cz; }

        float bv = -1.0f; int bi = 0;
        #pragma unroll
        for (int t = 0; t < NPTS / 256; ++t) {
            int i = tid + t * 256;
            float dx = px[i] - cx, dy = py[i] - cy, dz = pz[i] - cz;
            float d  = dx * dx + dy * dy + dz * dz;
            d = fminf(dist[i], d);
            dist[i] = d;
            if (d > bv) { bv = d; bi = i; }   // strict > keeps first (lowest i)
        }
        // wave32 shuffle argmax (tie -> lower index, matching jnp.argmax)
        #pragma unroll
        for (int off = 16; off > 0; off >>= 1) {
            float ov = __shfl_xor(bv, off, 32);
            int   oi = __shfl_xor(bi, off, 32);
            if (ov > bv || (ov == bv && oi < bi)) { bv = ov; bi = oi; }
        }
        const int w = tid >> 5;
        if ((tid & 31) == 0) { redV[w] = bv; redI[w] = bi; }
        __syncthreads();
        if (tid == 0) {
            float Bv = redV[0]; int Bi = redI[0];
            #pragma unroll
            for (int k = 1; k < 8; ++k)
                if (redV[k] > Bv || (redV[k] == Bv && redI[k] < Bi)) { Bv = redV[k]; Bi = redI[k]; }
            farS[0] = Bi;
        }
        __syncthreads();
        far = farS[0];
    }
}

// ---------------------------------------------------------------------------
// Kernel 2: Ball query. One wave32 per centroid; ballot/popc ordered append.
// ---------------------------------------------------------------------------
__global__ void ballq_kernel(const float* __restrict__ xyz,
                             const float* __restrict__ new_xyz,
                             int* __restrict__ idx_out)
{
    const int gwave = (blockIdx.x * blockDim.x + threadIdx.x) >> 5;
    const int lane  = threadIdx.x & 31;
    if (gwave >= BATCH * NPOINT) return;
    const int b = gwave >> 10;
    const int p = gwave & (NPOINT - 1);

    const float* c = new_xyz + (size_t)(b * NPOINT + p) * 3;
    const float cx = c[0], cy = c[1], cz = c[2];
    const float cc = cx * cx + cy * cy + cz * cz;
    int* out = idx_out + (size_t)(b * NPOINT + p) * NSAMPLE;
    const float* X = xyz + (size_t)b * NPTS * 3;

    int cnt = 0;
    int first = -1;
    for (int base = 0; base < NPTS && cnt < NSAMPLE; base += 32) {
        const int j = base + lane;
        const float x = X[j * 3 + 0], y = X[j * 3 + 1], z = X[j * 3 + 2];
        const float pp = x * x + y * y + z * z;
        const float d2 = cc + pp - 2.0f * (cx * x + cy * y + cz * z);  // same form as reference
        const bool in = d2 < 0.04f;                                   // RADIUS^2
        const unsigned int mask = (unsigned int)__ballot(in);
        if (mask) {
            if (first < 0) {
                const int low = __ffs((int)mask) - 1;
                first = __shfl(j, low, 32);
            }
            const int pos = __popc(mask & ((1u << lane) - 1u));
            if (in && (cnt + pos) < NSAMPLE) out[cnt + pos] = j;
            cnt += __popc(mask);
        }
    }
    if (cnt < NSAMPLE && lane >= cnt) out[lane] = first;  // pad with first in-ball index
}

// ---------------------------------------------------------------------------
// Fused gather + MLP (conv+BN+ReLU x3) + maxpool using fp32 WMMA 16x16x4.
// Each wave owns M-tile mt = wave (16 rows); NT output N-tiles per wave.
// Weights are staged in LDS in a k-pair-swizzled layout so each lane's B
// fragment (W[k][n], W[k+1][n]) is one aligned ds_load_b64 — no packing movs.
// ---------------------------------------------------------------------------
template <int NT, int K, int LDIN, int NW, int LDOUT>
__device__ __forceinline__ void mlp_layer_wmma(
    const float* __restrict__ Xin,
    const float* __restrict__ Wsw,          // swizzled: Wsw[kk*NW + n*2 + (k&1)]
    const float* __restrict__ g, const float* __restrict__ bb,
    const float* __restrict__ mn, const float* __restrict__ vr,
    float* __restrict__ Xout,
    int mt, int lane)
{
    const v8f zero = {0.f, 0.f, 0.f, 0.f, 0.f, 0.f, 0.f, 0.f};
    v8f acc[NT];
    #pragma unroll
    for (int t = 0; t < NT; ++t) acc[t] = zero;

    const int ml   = lane & 15;
    const int half = lane >> 4;
    const int arow = (mt * 16 + ml) * LDIN;

    #pragma unroll
    for (int ks = 0; ks < (K >> 2); ++ks) {
        const int kk = ks * 4 + half * 2;   // A 16x4 layout: vgpr0=K{0,2}, vgpr1=K{1,3}
        const v2f a = *(const v2f*)(Xin + arow + kk);          // single ds_load_b64
        #pragma unroll
        for (int t = 0; t < NT; ++t) {
            const int n = t * 16 + ml;
            const v2f bf = *(const v2f*)(Wsw + kk * NW + n * 2); // single ds_load_b64
            acc[t] = __builtin_amdgcn_wmma_f32_16x16x4_f32(
                false, a, false, bf, (short)0, acc[t], false, false);
        }
    }
    #pragma unroll
    for (int t = 0; t < NT; ++t) {
        const int n  = t * 16 + ml;
        const float sc = g[n] * rsqrtf(vr[n] + 1e-3f);
        const float sh = bb[n] - mn[n] * sc;
        #pragma unroll
        for (int r = 0; r < 8; ++r) {
            const int mrow = mt * 16 + r + 8 * half;  // C/D layout
            const float val = acc[t][r] * sc + sh;
            Xout[mrow * LDOUT + n] = fmaxf(val, 0.0f);
        }
    }
}

__global__ void group_mlp_kernel(
    const float* __restrict__ xyz, const float* __restrict__ points,
    const float* __restrict__ W0, const float* __restrict__ g0, const float* __restrict__ b0,
    const float* __restrict__ m0, const float* __restrict__ v0,
    const float* __restrict__ W1, const float* __restrict__ g1, const float* __restrict__ b1,
    const float* __restrict__ m1, const float* __restrict__ v1,
    const float* __restrict__ W2, const float* __restrict__ g2, const float* __restrict__ b2,
    const float* __restrict__ m2, const float* __restrict__ v2,
    const float* __restrict__ new_xyz, const int* __restrict__ idx,
    float* __restrict__ new_points)
{
    extern __shared__ float sm[];
    float* X0  = sm;                    // 128 x 68
    float* X1  = X0 + 128 * 68;         // 128 x 64
    float* X2  = X1 + 128 * 64;         // 128 x 64
    float* X3  = X2 + 128 * 64;         // 128 x 128
    float* W0s = X3 + 128 * 128;        // 68 x 64 swizzled (zero padded K rows)
    float* W1s = W0s + 68 * 64;         // 64 x 64 swizzled
    float* W2s = W1s + 64 * 64;         // 64 x 128 swizzled

    const int tid   = threadIdx.x;
    const int cbase = blockIdx.x * 4;   // 4 centroids per block
    const int b  = cbase >> 10;
    const int p0 = cbase & (NPOINT - 1);

    // Stage weights k-pair-swizzled: Wsw[(k>>1)*2*NW + n*2 + (k&1)]
    for (int i = tid; i < 68 * 64; i += 256) {
        const int k = i >> 6, n = i & 63;
        W0s[(k >> 1) * 128 + n * 2 + (k & 1)] = (k < 67) ? W0[i] : 0.0f;
    }
    for (int i = tid; i < 64 * 64; i += 256) {
        const int k = i >> 6, n = i & 63;
        W1s[(k >> 1) * 128 + n * 2 + (k & 1)] = W1[i];
    }
    for (int i = tid; i < 64 * 128; i += 256) {
        const int k = i >> 7, n = i & 127;
        W2s[(k >> 1) * 256 + n * 2 + (k & 1)] = W2[i];
    }

    // Gather: 128 rows of [relative xyz | 64 point feats | pad0]
    {
        const int r = tid & 127, hi = tid >> 7;
        const int c = r >> 5, s = r & 31;
        const int gp = b * NPOINT + p0 + c;
        const int j  = idx[(size_t)gp * NSAMPLE + s];
        const float* P  = points + ((size_t)b * NPTS + j) * CIN;
        float* row = X0 + r * 68;
        if (hi == 0) {
            const float* nx = new_xyz + (size_t)gp * 3;
            const float* pj = xyz + ((size_t)b * NPTS + j) * 3;
            row[0] = pj[0] - nx[0];
            row[1] = pj[1] - nx[1];
            row[2] = pj[2] - nx[2];
            #pragma unroll
            for (int t = 0; t < 32; ++t) row[3 + t] = P[t];
        } else {
            #pragma unroll
            for (int t = 32; t < 64; ++t) row[3 + t] = P[t];
            row[67] = 0.0f;
        }
    }
    __syncthreads();

    const int wave = tid >> 5, lane = tid & 31;
    mlp_layer_wmma<4, 68, 68, 64, 64>(X0, W0s, g0, b0, m0, v0, X1, wave, lane);
    __syncthreads();
    mlp_layer_wmma<4, 64, 64, 64, 64>(X1, W1s, g1, b1, m1, v1, X2, wave, lane);
    __syncthreads();
    mlp_layer_wmma<8, 64, 64, 128, 128>(X2, W2s, g2, b2, m2, v2, X3, wave, lane);
    __syncthreads();

    // Maxpool over the 32 samples of each centroid
    for (int o = tid; o < 4 * 128; o += 256) {
        const int c = o >> 7, d = o & 127;
        const float* col = X3 + (size_t)(c * 32) * 128 + d;
        float mx = col[0];
        #pragma unroll
        for (int s = 1; s < 32; ++s) mx = fmaxf(mx, col[s * 128]);
        new_points[((size_t)(b * NPOINT + p0 + c)) * 128 + d] = mx;
    }
}

// ---------------------------------------------------------------------------
extern "C" void kernel_launch(void* const* d_in, const int* in_sizes, int n_in,
                              void* d_out, int out_size, void* d_ws, size_t ws_size,
                              hipStream_t stream)
{
    const float* xyz    = (const float*)d_in[0];
    const float* points = (const float*)d_in[1];
    const float* W0 = (const float*)d_in[2];
    const float* g0 = (const float*)d_in[3];
    const float* b0 = (const float*)d_in[4];
    const float* m0 = (const float*)d_in[5];
    const float* v0 = (const float*)d_in[6];
    const float* W1 = (const float*)d_in[7];
    const float* g1 = (const float*)d_in[8];
    const float* b1 = (const float*)d_in[9];
    const float* m1 = (const float*)d_in[10];
    const float* v1 = (const float*)d_in[11];
    const float* W2 = (const float*)d_in[12];
    const float* g2 = (const float*)d_in[13];
    const float* b2 = (const float*)d_in[14];
    const float* m2 = (const float*)d_in[15];
    const float* v2 = (const float*)d_in[16];

    float* out        = (float*)d_out;
    float* new_xyz    = out;                                   // B*NPOINT*3
    float* new_points = out + BATCH * NPOINT * 3;              // B*NPOINT*128
    int*   idx        = (int*)(out + BATCH * NPOINT * 3 + BATCH * NPOINT * 128);

    const size_t fps_lds = (size_t)(4 * NPTS + 32) * sizeof(float);                    // ~65.7 KB
    const size_t mlp_lds = (size_t)(128 * 68 + 2 * 128 * 64 + 128 * 128
                                    + 68 * 64 + 64 * 64 + 64 * 128) * sizeof(float);   // ~232 KB

    (void)hipFuncSetAttribute((const void*)fps_kernel,
                              hipFuncAttributeMaxDynamicSharedMemorySize, (int)fps_lds);
    (void)hipFuncSetAttribute((const void*)group_mlp_kernel,
                              hipFuncAttributeMaxDynamicSharedMemorySize, (int)mlp_lds);

    // 1) FPS -> new_xyz
    fps_kernel<<<BATCH, 256, fps_lds, stream>>>(xyz, new_xyz);

    // 2) Ball query -> idx (one wave per centroid; 8 waves/block)
    ballq_kernel<<<(BATCH * NPOINT) / 8, 256, 0, stream>>>(xyz, new_xyz, idx);

    // 3) Fused gather + MLP(WMMA fp32) + maxpool -> new_points
    group_mlp_kernel<<<(BATCH * NPOINT) / 4, 256, mlp_lds, stream>>>(
        xyz, points,
        W0, g0, b0, m0, v0,
        W1, g1, b1, m1, v1,
        W2, g2, b2, m2, v2,
        new_xyz, idx, new_points);

    (void)in_sizes; (void)n_in; (void)out_size; (void)d_ws; (void)ws_size;
}